// PhysicsLossTransient_36919538876951
// MI455X (gfx1250) — compile-verified
//
#include <hip/hip_runtime.h>

typedef __attribute__((ext_vector_type(2))) float v2f;
typedef __attribute__((ext_vector_type(8))) float v8f;

#define NN 169
#define MT 176        // 11 tiles of 16 rows (padded)
#define KBSTR 49      // band row stride (48 cols + 1 pad -> conflict-free)
#define TTROWS 208    // T_old tile rows: node -16 .. 191
#define TSTR 17       // tile row stride (16 timesteps + 1 pad)
#define TNROWS 176    // T_new tile rows: node 0 .. 175

// float32-exact constants (match numpy/np.float32 build of K/E)
#define GLC 0.015f                       // thickness*board_k*(DY/DX) = 0.001*15
#define GRC ((float)(1.6/14400.0))       // 2*dx*dy*0.8, dx=dy=0.1/12
#define VHC 0.16875f                     // rho*cp*th*dx*dy = 2430/14400
#define SIGC 5.67e-8f

__device__ __forceinline__ float computeK(int m, int n) {
    if (m >= NN || n < 0 || n >= NN) return 0.0f;
    bool itf = (m == 0) | (m == 12) | (m == 156) | (m == 168);
    if (itf) return (m == n) ? 1.0f : 0.0f;
    int i = m % 13, j = m / 13;
    float v = 0.0f;
    if (n == m) {
        float d = 0.0f;
        if (i + 1 < 13) d += GLC;
        if (i - 1 >= 0) d += GLC;
        if (j + 1 < 13) d += GLC;
        if (j - 1 >= 0) d += GLC;
        v = d;
    } else if (n == m + 1 && i + 1 < 13)  v = -GLC;
    else if (n == m - 1 && i - 1 >= 0)    v = -GLC;
    else if (n == m + 13 && j + 1 < 13)   v = -GLC;
    else if (n == m - 13 && j - 1 >= 0)   v = -GLC;
    return v;
}

__global__ __launch_bounds__(128) void physloss_kernel(
    const float* __restrict__ Tpred, const float* __restrict__ Ttrue,
    const float* __restrict__ heat,  const float* __restrict__ itfv,
    const float* __restrict__ Tenv,  float* __restrict__ gsums)
{
    __shared__ float sKb[MT * KBSTR];     // banded K: sKb[m][c] = K[m, (m&~15)-16+c]
    __shared__ float sTt[TTROWS * TSTR];  // T_old transposed tile (node+16, t)
    __shared__ float sTn[TNROWS * TSTR];  // T_new transposed tile (node, t)
    __shared__ float sQ[MT];
    __shared__ float sE[MT];              // holds SIGMA * E (sigma folded in)
    __shared__ float sSum[16];

    const int tid   = threadIdx.x;
    const int b     = blockIdx.y;
    const int chunk = blockIdx.x;

    // ---- per-block init: banded K ----
    for (int idx = tid; idx < MT * 48; idx += 128) {
        int m = idx / 48, c = idx - m * 48;
        int col = (m & ~15) - 16 + c;
        sKb[m * KBSTR + c] = computeK(m, col);
    }
    // ---- Q and sigma*E vectors for this batch ----
    for (int m = tid; m < MT; m += 128) {
        bool isItf = (m == 0) | (m == 12) | (m == 156) | (m == 168);
        sE[m] = (m < NN && !isItf) ? (SIGC * GRC) : 0.0f;
        float q = 0.0f;
        if (m == 81)  q = heat[b * 4 + 0];
        if (m == 45)  q = heat[b * 4 + 1];
        if (m == 120) q = heat[b * 4 + 2];
        if (m == 126) q = heat[b * 4 + 3];
        if (m == 0)   q = itfv[b * 4 + 0];
        if (m == 12)  q = itfv[b * 4 + 1];
        if (m == 168) q = itfv[b * 4 + 2];
        if (m == 156) q = itfv[b * 4 + 3];
        sQ[m] = q;
    }
    // ---- static zero padding rows (stay zero across all groups) ----
    for (int idx = tid; idx < 16 * TSTR; idx += 128) sTt[idx] = 0.0f;                 // nodes -16..-1
    for (int idx = tid; idx < (TTROWS - 185) * TSTR; idx += 128) sTt[185 * TSTR + idx] = 0.0f; // 169..191
    for (int idx = tid; idx < (TNROWS - 169) * TSTR; idx += 128) sTn[169 * TSTR + idx] = 0.0f; // 169..175

    const int lane = tid & 31;
    const int wave = tid >> 5;
    const int half = lane >> 4;   // 0: lanes 0-15, 1: lanes 16-31
    const int tcol = lane & 15;   // timestep column held by this lane (C/D layout)

    const float te  = Tenv[b];
    const float te2 = te * te;
    const float te4 = te2 * te2;

    for (int g = 0; g < 4; ++g) {
        const int tbase = chunk * 64 + g * 16;
        const size_t slab0 = ((size_t)b * 512 + (size_t)tbase) * NN;  // 16B-aligned (tbase % 16 == 0)
        __syncthreads();   // previous group fully consumed

        if (tbase + 16 <= 511) {
            // ---- fast path: B128 loads of contiguous slabs, transpose-scatter into LDS ----
            // T_old: 2704 floats at slab0 (16B aligned)
            const float4* src4 = reinterpret_cast<const float4*>(Ttrue + slab0);
            for (int idx4 = tid; idx4 < 676; idx4 += 128) {
                float4 v4 = src4[idx4];
                float vv[4] = { v4.x, v4.y, v4.z, v4.w };
                int e = idx4 * 4;
                #pragma unroll
                for (int j = 0; j < 4; ++j) {
                    int ee = e + j;
                    int t = ee / NN, n = ee - t * NN;
                    sTt[(n + 16) * TSTR + t] = vv[j];
                }
            }
            // T_new: elements 169..2872 of the 17-row aligned region at slab0
            const float4* srp4 = reinterpret_cast<const float4*>(Tpred + slab0);
            for (int idx4 = 42 + tid; idx4 < 719; idx4 += 128) {
                float4 v4 = srp4[idx4];   // last one over-reads 3 floats, still in-bounds
                float vv[4] = { v4.x, v4.y, v4.z, v4.w };
                int e = idx4 * 4;
                #pragma unroll
                for (int j = 0; j < 4; ++j) {
                    int ee = e + j;
                    if (ee >= NN && ee < 17 * NN) {
                        int tt = ee / NN, n = ee - tt * NN;   // tt in 1..16 -> T_new t = tt-1
                        sTn[n * TSTR + (tt - 1)] = vv[j];
                    }
                }
            }
        } else {
            // ---- tail group (tbase = 496): guarded scalar loads ----
            for (int idx = tid; idx < 16 * NN; idx += 128) {
                int t = idx / NN, n = idx - t * NN;
                int tg = tbase + t;
                float to = 0.0f, tn = 0.0f;
                if (tg < 511) {
                    size_t base = slab0 + (size_t)t * NN + (size_t)n;
                    to = Ttrue[base];
                    tn = Tpred[base + NN];
                }
                sTt[(n + 16) * TSTR + t] = to;
                sTn[n * TSTR + t]        = tn;
            }
        }
        if (tid < 16) sSum[tid] = 0.0f;

        // prefetch next group's slabs while this group computes
        if (g < 3) {
            const float* pn0 = Ttrue + slab0 + 16 * NN + tid * 22;
            const float* pn1 = Tpred + slab0 + 17 * NN + tid * 22;
            __builtin_prefetch(pn0, 0, 1);
            __builtin_prefetch(pn1, 0, 1);
        }
        __syncthreads();

        // ---- banded GEMM via V_WMMA_F32_16X16X4_F32 + fused residual readout ----
        float s = 0.0f;
        for (int mtile = wave; mtile < 11; mtile += 4) {
            v8f acc = {};
            const int mA = mtile * 16 + (lane & 15);   // A row held by this lane
            #pragma unroll
            for (int kk = 0; kk < 12; ++kk) {
                // A fragment: VGPR0 = K[m, kb+0]/K[m, kb+2], VGPR1 = +1/+3
                int c = kk * 4 + half * 2;
                v2f a;
                a.x = sKb[mA * KBSTR + c];
                a.y = sKb[mA * KBSTR + c + 1];
                // B fragment: rows k striped across lanes; row_in_sTt = kglobal + 16
                int row = mtile * 16 + kk * 4 + half * 2;
                v2f bb;
                bb.x = sTt[row * TSTR + tcol];
                bb.y = sTt[(row + 1) * TSTR + tcol];
                acc = __builtin_amdgcn_wmma_f32_16x16x4_f32(
                        false, a, false, bb, (short)0, acc, false, false);
            }
            // D layout: VGPR r -> row mtile*16 + r + half*8, column tcol
            #pragma unroll
            for (int r = 0; r < 8; ++r) {
                int m = mtile * 16 + r + half * 8;
                float told = sTt[(m + 16) * TSTR + tcol];
                float tnew = sTn[m * TSTR + tcol];
                float t2 = told * told;
                float res = VHC * (tnew - told) - sQ[m] + acc[r]
                          + sE[m] * (t2 * t2 - te4);
                s += fabsf(res);
            }
        }
        // per-timestep partial sums (rows >=169 contributed exact zeros)
        if (tbase + tcol < 511) atomicAdd(&sSum[tcol], s);
        __syncthreads();
        if (tid < 16 && tbase + tid < 511) atomicAdd(&gsums[tbase + tid], sSum[tid]);
    }
}

__global__ __launch_bounds__(512) void physloss_finalize(
    const float* __restrict__ gsums, float* __restrict__ out)
{
    __shared__ float red[512];
    int t = threadIdx.x;
    float v = 0.0f;
    if (t < 511) {
        float lt = gsums[t] * (1.0f / 43264.0f);   // mean over B*NN = 256*169
        v = lt * lt;
    }
    red[t] = v;
    __syncthreads();
    for (int s = 256; s > 0; s >>= 1) {
        if (t < s) red[t] += red[t + s];
        __syncthreads();
    }
    if (t == 0) out[0] = red[0] * (1.0f / 511.0f);
}

extern "C" void kernel_launch(void* const* d_in, const int* in_sizes, int n_in,
                              void* d_out, int out_size, void* d_ws, size_t ws_size,
                              hipStream_t stream) {
    const float* Tpred = (const float*)d_in[0];   // [256,512,1,13,13]
    const float* Ttrue = (const float*)d_in[1];   // [256,512,1,13,13]
    const float* heat  = (const float*)d_in[2];   // [256,4]
    const float* itfv  = (const float*)d_in[3];   // [256,4]
    const float* Tenv  = (const float*)d_in[4];   // [256,1]
    float* gsums = (float*)d_ws;                  // 511 per-timestep |res| sums
    float* out   = (float*)d_out;

    hipMemsetAsync(gsums, 0, 511 * sizeof(float), stream);

    dim3 grid(8, 256);   // 8 timestep-chunks (64 t each) x 256 batches
    dim3 block(128);     // 4 wave32
    physloss_kernel<<<grid, block, 0, stream>>>(Tpred, Ttrue, heat, itfv, Tenv, gsums);
    physloss_finalize<<<1, 512, 0, stream>>>(gsums, out);
}